// EmbeddingLayer_29008209117742
// MI455X (gfx1250) — compile-verified
//
#include <hip/hip_runtime.h>

typedef __attribute__((ext_vector_type(2))) float v2f;
typedef __attribute__((ext_vector_type(8))) float v8f;

#define NEG_SLOPE 0.01f
static constexpr int kB = 2;
static constexpr int kN = 50000;
static constexpr int kD = 128;

// ---------------------------------------------------------------- zero ws
__global__ void zero_ws_kernel(float4* __restrict__ ws, int n4) {
  int i = blockIdx.x * blockDim.x + threadIdx.x;
  const int stride = gridDim.x * blockDim.x;
  float4 z; z.x = 0.f; z.y = 0.f; z.z = 0.f; z.w = 0.f;
  for (; i < n4; i += stride) ws[i] = z;
}

// ---------------------------------------------------------------- scatter
// One wave32 per edge. Lane l owns D-slice [4l, 4l+4). For each batch:
//   nbr[b,u,:] += prev[b,v,:]   and   nbr[b,v,:] += prev[b,u,:]
__global__ void __launch_bounds__(256) scatter_kernel(
    const float* __restrict__ prev, const int* __restrict__ edges,
    float* __restrict__ nbr, int E) {
  const int wave = (blockIdx.x * blockDim.x + threadIdx.x) >> 5;
  const int lane = threadIdx.x & 31;
  if (wave >= E) return;
  const int u = edges[2 * wave + 0];
  const int v = edges[2 * wave + 1];
  const int d0 = lane * 4;
#pragma unroll
  for (int b = 0; b < kB; ++b) {
    const size_t ru = ((size_t)b * kN + u) * kD + d0;
    const size_t rv = ((size_t)b * kN + v) * kD + d0;
    float4 pv = *(const float4*)(prev + rv);
    float4 pu = *(const float4*)(prev + ru);
    atomicAdd(nbr + ru + 0, pv.x);
    atomicAdd(nbr + ru + 1, pv.y);
    atomicAdd(nbr + ru + 2, pv.z);
    atomicAdd(nbr + ru + 3, pv.w);
    atomicAdd(nbr + rv + 0, pu.x);
    atomicAdd(nbr + rv + 1, pu.y);
    atomicAdd(nbr + rv + 2, pu.z);
    atomicAdd(nbr + rv + 3, pu.w);
  }
}

// ---------------------------------------------------------------- GEMM + bias + leaky relu
// out[r, e] = lrelu(node[r,e] + edge[r,e] + sum_d nbr[r,d] * W2[e,d])
// Block: 256 threads = 8 waves. Block handles 16 rows; wave w handles the
// 16x16 tile at columns [16w, 16w+16). K-loop: 32 x v_wmma_f32_16x16x4_f32.
//
// f32 16x4 A layout (ISA 7.12.2): lane L holds A[L%16, k0] in reg0 and
// A[L%16, k0+1] in reg1, with k0 = (L<16 ? 0 : 2). B (4x16) is symmetric with
// lane L holding column L%16, so a lane loads W2[e0 + L%16, k0..k0+1].
__global__ void __launch_bounds__(256) gemm_lrelu_kernel(
    const float* __restrict__ nbr, const float* __restrict__ W2,
    const float* __restrict__ nodef, const float* __restrict__ edgef,
    float* __restrict__ out) {
  const int wave = threadIdx.x >> 5;   // 0..7 -> e tile
  const int lane = threadIdx.x & 31;
  const int half = lane >> 4;          // 0 or 1
  const int l16  = lane & 15;
  const int row0 = blockIdx.x * 16;    // tile base in flattened (b*N + n)
  const int e0   = wave * 16;

  const float* Arow = nbr + (size_t)(row0 + l16) * kD + half * 2;
  const float* Brow = W2  + (size_t)(e0   + l16) * kD + half * 2;

  v8f acc = {0.f, 0.f, 0.f, 0.f, 0.f, 0.f, 0.f, 0.f};
#pragma unroll
  for (int k = 0; k < kD; k += 4) {
    v2f a = *(const v2f*)(Arow + k);
    v2f b = *(const v2f*)(Brow + k);
    acc = __builtin_amdgcn_wmma_f32_16x16x4_f32(
        /*neg_a=*/false, a, /*neg_b=*/false, b,
        /*c_mod=*/(short)0, acc, /*reuse_a=*/false, /*reuse_b=*/false);
  }

  // C/D layout: acc[i] = D[M = i + 8*half, N = l16]
#pragma unroll
  for (int i = 0; i < 8; ++i) {
    const size_t idx = (size_t)(row0 + i + 8 * half) * kD + e0 + l16;
    float s = nodef[idx] + edgef[idx] + acc[i];
    out[idx] = (s >= 0.f) ? s : NEG_SLOPE * s;
  }
}

// ---------------------------------------------------------------- launch
extern "C" void kernel_launch(void* const* d_in, const int* in_sizes, int n_in,
                              void* d_out, int out_size, void* d_ws, size_t ws_size,
                              hipStream_t stream) {
  const float* prev  = (const float*)d_in[0];
  const int*   edges = (const int*)d_in[1];
  const float* nodef = (const float*)d_in[2];
  const float* edgef = (const float*)d_in[3];
  const float* W2    = (const float*)d_in[4];
  float* out = (float*)d_out;
  float* nbr = (float*)d_ws;                  // B*N*D f32 = 51.2 MB accumulator

  const int E = in_sizes[1] / 2;
  const int total = kB * kN * kD;             // 12.8M floats
  const int n4 = total / 4;

  zero_ws_kernel<<<2048, 256, 0, stream>>>((float4*)nbr, n4);

  const int scatter_blocks = (E * 32 + 255) / 256;   // one wave per edge
  scatter_kernel<<<scatter_blocks, 256, 0, stream>>>(prev, edges, nbr, E);

  const int rows = kB * kN;                   // 100000, divisible by 16
  gemm_lrelu_kernel<<<rows / 16, 256, 0, stream>>>(nbr, W2, nodef, edgef, out);
}